// LoraLinear_10831907521048
// MI455X (gfx1250) — compile-verified
//
#include <hip/hip_runtime.h>

// ---------------------------------------------------------------------------
// LoRA linear for MI455X (gfx1250, wave32).
// Memory-bound (~450MB HBM vs ~275 GFLOP -> >10x memory-bound at 23.3 TB/s),
// so keep full fp32 precision and use V_WMMA_F32_16X16X4_F32 for all GEMMs.
// Kernel 2 streams K-tiles with GLOBAL_LOAD_ASYNC_TO_LDS_B128 (ASYNCcnt) into
// double-buffered LDS so loads overlap the WMMA pipeline.
// ---------------------------------------------------------------------------

typedef __attribute__((ext_vector_type(2))) float v2f;
typedef __attribute__((ext_vector_type(4))) float v4f;
typedef __attribute__((ext_vector_type(8))) float v8f;
typedef __attribute__((ext_vector_type(4))) int   v4i;

// D = A(16x4 f32) * B(4x16 f32) + C(16x16 f32)
#define WMMA4(a, b, c) \
  __builtin_amdgcn_wmma_f32_16x16x4_f32(false, (a), false, (b), (short)0, (c), false, false)

#define RLORA 16
#define DROPOUT_KEEP 0.95f
#define LORA_SCALE 2.0f  // alpha / r = 32 / 16

#if defined(__AMDGCN__) && __has_builtin(__builtin_amdgcn_global_load_async_to_lds_b128)
#define USE_ASYNC 1
#else
#define USE_ASYNC 0
#endif

// Copy 16B global -> LDS for this lane. Async path: memory->LDS DMA, no VGPR
// round trip, tracked by ASYNCcnt. Fallback: plain load + ds_store.
__device__ __forceinline__ void copy16(const float* __restrict__ g, float* l) {
#if USE_ASYNC
  // builtin signature (from hipcc diagnostic): (v4i AS1*, v4i AS3*, imm, imm)
  __builtin_amdgcn_global_load_async_to_lds_b128(
      (__attribute__((address_space(1))) v4i*)(void*)g,
      (__attribute__((address_space(3))) v4i*)(void*)l,
      0, 0);
#else
  *(v4f*)l = *(const v4f*)g;
#endif
}

__device__ __forceinline__ void wait_copies() {
#if USE_ASYNC
#if __has_builtin(__builtin_amdgcn_s_wait_asynccnt)
  __builtin_amdgcn_s_wait_asynccnt(0);
#else
  asm volatile("s_wait_asynccnt 0" ::: "memory");
#endif
#endif
}

// ---------------------------------------------------------------------------
// Kernel 1: XA[m, r] = (LORA_SCALE / DROPOUT_KEEP) * sum_k x[m,k]*mask[m,k]*A[r,k]
// One wave per 16-row tile; N = r = 16; K-loop of f32 WMMA.
// A-frag (16x4 f32): lane l -> row l%16, K = kbase + 2*(l/16) + {0,1}
// B-frag (4x16 f32): lane l -> col l%16, K = kbase + 2*(l/16) + {0,1}
// ---------------------------------------------------------------------------
__global__ __launch_bounds__(256) void lora_xa_kernel(
    const float* __restrict__ X, const unsigned char* __restrict__ Mask,
    const float* __restrict__ A, float* __restrict__ XA, int K)
{
  const int lane = threadIdx.x & 31;
  const int wave = threadIdx.x >> 5;
  const int lr   = lane & 15;
  const int hi   = lane >> 4;
  const int m0   = (blockIdx.x * 8 + wave) * 16;

  const float*         xrow = X    + (size_t)(m0 + lr) * K;
  const unsigned char* mrow = Mask + (size_t)(m0 + lr) * K;
  const float*         arow = A    + (size_t)lr * K;

  v8f acc = {};
#pragma unroll 4
  for (int k = 0; k < K; k += 4) {
    const int kk = k + 2 * hi;
    v2f a = *(const v2f*)(xrow + kk);
    // mask is {0,1}: byte->float convert + multiply (v_cvt_f32_ubyte0/1)
    const unsigned int ms = *(const unsigned short*)(mrow + kk);
    a.x *= (float)(ms & 0xffu);
    a.y *= (float)(ms >> 8);
    v2f b = *(const v2f*)(arow + kk);  // B_wmma[k][r] = A[r][k], contiguous 8B
    acc = WMMA4(a, b, acc);
  }

  const float s = LORA_SCALE / DROPOUT_KEEP;
#pragma unroll
  for (int v = 0; v < 8; ++v)
    XA[(size_t)(m0 + v + 8 * hi) * RLORA + lr] = acc[v] * s;
}

// ---------------------------------------------------------------------------
// Kernel 2: Out = X @ W^T + XA @ Bl^T
// Workgroup = 256 threads (8 waves), tile 128(M) x 128(N), KB = 32.
// Waves 4(M) x 2(N): each wave owns 32x64 = 2x4 v8f accumulators.
// LDS stride padded to 36 floats: 16B-aligned b128 fills, conflict-free
// float2 fragment reads (36*r mod 64 distinct for 16 rows).
// Double-buffered async staging: wait(cur) -> barrier -> prefetch(next) ->
// compute(cur). Rank-16 LoRA term folded into the same accumulators.
// ---------------------------------------------------------------------------
#define TM 128
#define TN 128
#define KB 32
#define LSTR 36

__global__ __launch_bounds__(256) void lora_main_kernel(
    const float* __restrict__ X, const float* __restrict__ W,
    const float* __restrict__ Bl, const float* __restrict__ XA,
    float* __restrict__ Out, int N, int K)
{
  __shared__ float xs[2][TM * LSTR];
  __shared__ float ws[2][TN * LSTR];

  const int t    = threadIdx.x;
  const int lane = t & 31;
  const int wave = t >> 5;
  const int wm   = wave & 3;   // 0..3 : 32-row band
  const int wn   = wave >> 2;  // 0..1 : 64-col band
  const int lr   = lane & 15;
  const int hi   = lane >> 4;

  const int tile_m = blockIdx.y * TM;
  const int tile_n = blockIdx.x * TN;

  // staging map: 8 threads cover one row's 32 floats (128B, coalesced)
  const int lrow = t >> 3;         // 0..31
  const int lcol = (t & 7) * 4;    // 0,4,...,28

  const float* xg = X + (size_t)(tile_m + lrow) * K + lcol;
  const float* wg = W + (size_t)(tile_n + lrow) * K + lcol;

  v8f acc[2][4] = {};

  const int NK = K / KB;

  // prologue: stage tile 0 into buffer 0
#pragma unroll
  for (int i = 0; i < 4; ++i) {
    copy16(xg + (size_t)(32 * i) * K, &xs[0][(lrow + 32 * i) * LSTR + lcol]);
    copy16(wg + (size_t)(32 * i) * K, &ws[0][(lrow + 32 * i) * LSTR + lcol]);
  }

  for (int kt = 0; kt < NK; ++kt) {
    const int cur = kt & 1;
    wait_copies();      // my async copies into buffer `cur` have landed
    __syncthreads();    // everyone's copies landed; buffer cur^1 free to reuse

    if (kt + 1 < NK) {
      const int k0 = (kt + 1) * KB;
#pragma unroll
      for (int i = 0; i < 4; ++i) {
        copy16(xg + (size_t)(32 * i) * K + k0,
               &xs[cur ^ 1][(lrow + 32 * i) * LSTR + lcol]);
        copy16(wg + (size_t)(32 * i) * K + k0,
               &ws[cur ^ 1][(lrow + 32 * i) * LSTR + lcol]);
      }
    }

#pragma unroll
    for (int kc = 0; kc < KB; kc += 4) {
      const int kk = kc + 2 * hi;
      v2f a0 = *(const v2f*)(&xs[cur][(wm * 32 +      lr) * LSTR + kk]);
      v2f a1 = *(const v2f*)(&xs[cur][(wm * 32 + 16 + lr) * LSTR + kk]);
#pragma unroll
      for (int j = 0; j < 4; ++j) {
        v2f b = *(const v2f*)(&ws[cur][(wn * 64 + 16 * j + lr) * LSTR + kk]);
        acc[0][j] = WMMA4(a0, b, acc[0][j]);
        acc[1][j] = WMMA4(a1, b, acc[1][j]);
      }
    }
  }

  // LoRA rank-16 epilogue (XA already carries dropout + alpha/r scaling).
  // XA (512KB) and Bl (256KB) are L2-resident; read fragments from global.
#pragma unroll
  for (int kc = 0; kc < RLORA; kc += 4) {
    const int kk = kc + 2 * hi;
    v2f a0 = *(const v2f*)(XA + (size_t)(tile_m + wm * 32 +      lr) * RLORA + kk);
    v2f a1 = *(const v2f*)(XA + (size_t)(tile_m + wm * 32 + 16 + lr) * RLORA + kk);
#pragma unroll
    for (int j = 0; j < 4; ++j) {
      v2f b = *(const v2f*)(Bl + (size_t)(tile_n + wn * 64 + 16 * j + lr) * RLORA + kk);
      acc[0][j] = WMMA4(a0, b, acc[0][j]);
      acc[1][j] = WMMA4(a1, b, acc[1][j]);
    }
  }

  // Store. C/D layout: VGPR v -> row = v + 8*(lane/16), col = lane%16.
  // Lanes 0-15 of each half-wave write 64B contiguous segments per row.
#pragma unroll
  for (int i = 0; i < 2; ++i) {
#pragma unroll
    for (int j = 0; j < 4; ++j) {
      const int rbase = tile_m + wm * 32 + 16 * i + 8 * hi;
      const int col   = tile_n + wn * 64 + 16 * j + lr;
#pragma unroll
      for (int v = 0; v < 8; ++v)
        Out[(size_t)(rbase + v) * N + col] = acc[i][j][v];
    }
  }
}

// ---------------------------------------------------------------------------
extern "C" void kernel_launch(void* const* d_in, const int* in_sizes, int n_in,
                              void* d_out, int out_size, void* d_ws, size_t ws_size,
                              hipStream_t stream) {
  (void)n_in; (void)out_size; (void)ws_size;
  const float*         X    = (const float*)d_in[0];        // [M, K]
  const float*         W    = (const float*)d_in[1];        // [N, K]
  const float*         A    = (const float*)d_in[2];        // [16, K]
  const float*         Bl   = (const float*)d_in[3];        // [N, 16]
  const unsigned char* Mask = (const unsigned char*)d_in[4];// [M, K] bool

  const int K = in_sizes[2] / RLORA;  // 4096
  const int M = in_sizes[0] / K;      // 8192
  const int N = in_sizes[3] / RLORA;  // 4096

  float* XA = (float*)d_ws;           // [M, 16] fp32 scratch (512 KB)

  lora_xa_kernel<<<dim3(M / (16 * 8)), dim3(256), 0, stream>>>(X, Mask, A, XA, K);
  lora_main_kernel<<<dim3(N / TN, M / TM), dim3(256), 0, stream>>>(
      X, W, Bl, XA, (float*)d_out, N, K);
}